// MambaSKALM_61512521614005
// MI455X (gfx1250) — compile-verified
//
#include <hip/hip_runtime.h>
#include <math.h>

// ---------------------------------------------------------------------------
// MambaSKALM on MI455X (gfx1250): fp32 WMMA pipeline.
//   B=4 T=4096 D=1024 H=16 R=32 P=64, POWER_K=2, N_PI=6, RIDGE=1e-3
// ---------------------------------------------------------------------------

typedef __attribute__((ext_vector_type(2))) float v2f;
typedef __attribute__((ext_vector_type(8))) float v8f;

#define T_DIM 4096
#define D_DIM 1024
#define H_DIM 16
#define R_DIM 32
#define P_DIM 64
#define RIDGE_C 1e-3f

// D = A(16x4) * B(4x16) + C, fp32 WMMA (CDNA5)
__device__ __forceinline__ v8f wmma4(v2f a, v2f b, v8f c) {
  return __builtin_amdgcn_wmma_f32_16x16x4_f32(
      /*neg_a=*/false, a, /*neg_b=*/false, b,
      /*c_mod=*/(short)0, c, /*reuse_a=*/false, /*reuse_b=*/false);
}

// ---------------------------------------------------------------------------
// Kernel 1: z/zq/v = h @ [Wk|Wq|Wv]^T   (M=B*T=16384, K=1024, N=2048)
// Block: 256 thr (8 waves), 128 rows x 64 cols; wave = 32x32 (4 WMMA tiles).
// h (64MB) fits in the 192MB L2, so column-block re-reads hit L2 not HBM.
// ---------------------------------------------------------------------------
__global__ __launch_bounds__(256)
void qkv_gemm_kernel(const float* __restrict__ h,
                     const float* __restrict__ Wk,
                     const float* __restrict__ Wq,
                     const float* __restrict__ Wv,
                     float* __restrict__ z, float* __restrict__ zq,
                     float* __restrict__ vout) {
  __shared__ float hT[128][20];   // padded: 16B-aligned rows, conflict-free
  __shared__ float wT[16][64];
  const int tid  = threadIdx.x;
  const int lane = tid & 31;
  const int wave = tid >> 5;
  const int rowBlock = blockIdx.y * 128;
  const int colBlock = blockIdx.x * 64;

  const float* W; float* outp; int cb; int outW;
  if (colBlock < 512)       { W = Wk; outp = z;    cb = colBlock;        outW = 512;  }
  else if (colBlock < 1024) { W = Wq; outp = zq;   cb = colBlock - 512;  outW = 512;  }
  else                      { W = Wv; outp = vout; cb = colBlock - 1024; outW = 1024; }

  const int wr = (wave >> 1) * 32;   // wave row offset: 0,32,64,96
  const int wc = (wave & 1) * 32;    // wave col offset: 0,32
  const int m    = lane & 15;
  const int half = lane >> 4;
  const int kb   = half * 2;

  v8f acc00 = {}, acc01 = {}, acc10 = {}, acc11 = {};

  const int hr = tid >> 1;           // 0..127
  const int hc = (tid & 1) * 8;      // 0 or 8
  const float* hsrc = h + (size_t)(rowBlock + hr) * D_DIM + hc;
  const int wci = tid >> 2;          // 0..63
  const int wk4 = (tid & 3) * 4;     // 0,4,8,12
  const float* wsrc = W + (size_t)(cb + wci) * D_DIM + wk4;

  for (int k0 = 0; k0 < D_DIM; k0 += 16) {
    float4 a0 = *(const float4*)(hsrc + k0);
    float4 a1 = *(const float4*)(hsrc + k0 + 4);
    float4 w4 = *(const float4*)(wsrc + k0);
    if (k0 + 16 < D_DIM) {
      __builtin_prefetch(hsrc + k0 + 16, 0, 0);   // global_prefetch_b8
      __builtin_prefetch(wsrc + k0 + 16, 0, 0);
    }
    *(float4*)&hT[hr][hc]     = a0;
    *(float4*)&hT[hr][hc + 4] = a1;
    wT[wk4 + 0][wci] = w4.x;
    wT[wk4 + 1][wci] = w4.y;
    wT[wk4 + 2][wci] = w4.z;
    wT[wk4 + 3][wci] = w4.w;
    __syncthreads();
#pragma unroll
    for (int ks = 0; ks < 16; ks += 4) {
      v2f fa0, fa1, fb0, fb1;
      fa0[0] = hT[wr + m][ks + kb];       fa0[1] = hT[wr + m][ks + kb + 1];
      fa1[0] = hT[wr + 16 + m][ks + kb];  fa1[1] = hT[wr + 16 + m][ks + kb + 1];
      fb0[0] = wT[ks + kb][wc + m];       fb0[1] = wT[ks + kb + 1][wc + m];
      fb1[0] = wT[ks + kb][wc + 16 + m];  fb1[1] = wT[ks + kb + 1][wc + 16 + m];
      acc00 = wmma4(fa0, fb0, acc00);
      acc01 = wmma4(fa0, fb1, acc01);
      acc10 = wmma4(fa1, fb0, acc10);
      acc11 = wmma4(fa1, fb1, acc11);
    }
    __syncthreads();
  }
  // C layout: VGPR j -> row j + 8*half; col = lane&15
#pragma unroll
  for (int jv = 0; jv < 8; ++jv) {
    size_t r0 = (size_t)(rowBlock + wr + 8 * half + jv);
    size_t r1 = r0 + 16;
    outp[r0 * outW + (cb + wc + m)]      = acc00[jv];
    outp[r0 * outW + (cb + wc + 16 + m)] = acc01[jv];
    outp[r1 * outW + (cb + wc + m)]      = acc10[jv];
    outp[r1 * outW + (cb + wc + 16 + m)] = acc11[jv];
  }
}

// ---------------------------------------------------------------------------
// Kernel 2: invmax[b,h] = 1 / max(max_t ||z[b,t,h,:]||, 1e-6)
// ---------------------------------------------------------------------------
__global__ __launch_bounds__(256)
void maxnorm_kernel(const float* __restrict__ z, float* __restrict__ invmax) {
  __shared__ float red[256];
  const int bh = blockIdx.x;
  const int b = bh >> 4, hh = bh & 15;
  const float* zp = z + (size_t)b * T_DIM * (H_DIM * R_DIM) + hh * R_DIM;
  float mx = 0.f;
  for (int t = threadIdx.x; t < T_DIM; t += 256) {
    const float* p = zp + (size_t)t * 512;
    float s = 0.f;
#pragma unroll
    for (int r = 0; r < 32; r += 4) {
      float4 q = *(const float4*)(p + r);
      s += q.x * q.x + q.y * q.y + q.z * q.z + q.w * q.w;
    }
    mx = fmaxf(mx, s);
  }
  red[threadIdx.x] = mx;
  __syncthreads();
  for (int s = 128; s > 0; s >>= 1) {
    if (threadIdx.x < s) red[threadIdx.x] = fmaxf(red[threadIdx.x], red[threadIdx.x + s]);
    __syncthreads();
  }
  if (threadIdx.x == 0) {
    float mn = fmaxf(sqrtf(red[0]), 1e-6f);
    invmax[bh] = 1.0f / mn;
  }
}

// ---------------------------------------------------------------------------
// Kernel 3: per (b,h) masked Grams via WMMA over T:
//   G  = sum_t (m z)(m z)^T           (32x32, raw z; 1/max^2 folded later)
//   M  = sum_t (ml z_{t+1})(ml z_t)^T (32x32)
//   Cv = sum_t (m v)(m z)^T           (64x32)
// 8 waves each own disjoint 16-token chunks; LDS ds_add reduction at the end.
// ---------------------------------------------------------------------------
__global__ __launch_bounds__(256)
void gram_kernel(const float* __restrict__ z, const float* __restrict__ vv_,
                 const float* __restrict__ pmask,
                 float* __restrict__ Gout, float* __restrict__ Mout,
                 float* __restrict__ Cvout) {
  __shared__ float sG[1024];
  __shared__ float sM[1024];
  __shared__ float sC[2048];
  const int bh = blockIdx.x;
  const int b = bh >> 4, hh = bh & 15;
  const int tid = threadIdx.x, lane = tid & 31, wave = tid >> 5;
  const float* zp = z   + (size_t)b * T_DIM * 512  + hh * R_DIM;
  const float* vp = vv_ + (size_t)b * T_DIM * 1024 + hh * P_DIM;
  const float* pm = pmask + (size_t)b * T_DIM;

  for (int i = tid; i < 1024; i += 256) { sG[i] = 0.f; sM[i] = 0.f; }
  for (int i = tid; i < 2048; i += 256) sC[i] = 0.f;
  __syncthreads();

  const int m = lane & 15, half = lane >> 4, kb = half * 2;
  v8f aG[2][2] = {}; v8f aM[2][2] = {}; v8f aC[4][2] = {};

  for (int t0 = wave * 16; t0 < T_DIM; t0 += 128) {
#pragma unroll
    for (int ks = 0; ks < 16; ks += 4) {
      const int tx = t0 + ks + kb;          // even token of this lane's k-pair
      const int ty = tx + 1;
      const float mx  = pm[tx], my = pm[ty];
      const float mlx = mx * my;            // lag mask for pair (tx, tx+1)
      const int   tyn = (ty + 1 < T_DIM) ? (ty + 1) : (T_DIM - 1);
      const float mly = (ty + 1 < T_DIM) ? my * pm[ty + 1] : 0.f;

      const float* zx = zp + (size_t)tx * 512;
      const float* zy = zp + (size_t)ty * 512;
      const float* zg = zp + (size_t)tyn * 512;
      float zx0 = zx[m], zx1 = zx[16 + m];
      float zy0 = zy[m], zy1 = zy[16 + m];
      float zg0 = zg[m], zg1 = zg[16 + m];

      v2f f0; f0[0] = zx0 * mx;  f0[1] = zy0 * my;   // z*m  cols 0-15
      v2f f1; f1[0] = zx1 * mx;  f1[1] = zy1 * my;   // z*m  cols 16-31
      v2f e0; e0[0] = zx0 * mlx; e0[1] = zy0 * mly;  // z_t * mlag
      v2f e1; e1[0] = zx1 * mlx; e1[1] = zy1 * mly;
      v2f g0; g0[0] = zy0 * mlx; g0[1] = zg0 * mly;  // z_{t+1} * mlag
      v2f g1; g1[0] = zy1 * mlx; g1[1] = zg1 * mly;

      aG[0][0] = wmma4(f0, f0, aG[0][0]); aG[0][1] = wmma4(f0, f1, aG[0][1]);
      aG[1][0] = wmma4(f1, f0, aG[1][0]); aG[1][1] = wmma4(f1, f1, aG[1][1]);
      aM[0][0] = wmma4(g0, e0, aM[0][0]); aM[0][1] = wmma4(g0, e1, aM[0][1]);
      aM[1][0] = wmma4(g1, e0, aM[1][0]); aM[1][1] = wmma4(g1, e1, aM[1][1]);

      const float* vx = vp + (size_t)tx * 1024;
      const float* vy = vp + (size_t)ty * 1024;
#pragma unroll
      for (int pi = 0; pi < 4; ++pi) {
        v2f u; u[0] = vx[pi * 16 + m] * mx; u[1] = vy[pi * 16 + m] * my;
        aC[pi][0] = wmma4(u, f0, aC[pi][0]);
        aC[pi][1] = wmma4(u, f1, aC[pi][1]);
      }
    }
  }
  // cross-wave reduction through LDS atomics (ds_add_f32)
#pragma unroll
  for (int i = 0; i < 2; ++i)
#pragma unroll
    for (int j = 0; j < 2; ++j)
#pragma unroll
      for (int jv = 0; jv < 8; ++jv) {
        int row = i * 16 + jv + 8 * half;
        int col = j * 16 + m;
        atomicAdd(&sG[row * 32 + col], aG[i][j][jv]);
        atomicAdd(&sM[row * 32 + col], aM[i][j][jv]);
      }
#pragma unroll
  for (int i = 0; i < 4; ++i)
#pragma unroll
    for (int j = 0; j < 2; ++j)
#pragma unroll
      for (int jv = 0; jv < 8; ++jv) {
        int row = i * 16 + jv + 8 * half;
        int col = j * 16 + m;
        atomicAdd(&sC[row * 32 + col], aC[i][j][jv]);
      }
  __syncthreads();
  float* Gp = Gout  + (size_t)bh * 1024;
  float* Mp = Mout  + (size_t)bh * 1024;
  float* Cp = Cvout + (size_t)bh * 2048;
  for (int i = tid; i < 1024; i += 256) { Gp[i] = sG[i]; Mp[i] = sM[i]; }
  for (int i = tid; i < 2048; i += 256) Cp[i] = sC[i];
}

// ---------------------------------------------------------------------------
// Kernel 4: per (b,h), one wave: Cholesky, L^-1, A_w = L^-1 M L^-T, power
// iteration (6), spectral clamp * gamma, B_v = Cv G^-1, and fold the whole
// output operator into  E^T = (eta/maxnorm) * (B_v L A_w A_w L^-1)^T  (32x64)
// ---------------------------------------------------------------------------
__global__ __launch_bounds__(32)
void solve_kernel(const float* __restrict__ Gin, const float* __restrict__ Min,
                  const float* __restrict__ Cin, const float* __restrict__ invmax,
                  const float* __restrict__ eta_p, const float* __restrict__ gamma_p,
                  float* __restrict__ Ert) {
  __shared__ float Lm[32][32], Li[32][32], Mm[32][32], Tt[32][32], Aw[32][32];
  __shared__ float Gi[32][32], S1[32][32], S2[32][32], S3[32][32];
  __shared__ float Cv[64][32], Bv[64][32];
  __shared__ float vec[32], uu[32], red[32];
  const int bh = blockIdx.x;
  const int l  = threadIdx.x;
  const float inv  = invmax[bh];
  const float inv2 = inv * inv;
  const float* Gp = Gin + (size_t)bh * 1024;
  const float* Mp = Min + (size_t)bh * 1024;
  const float* Cp = Cin + (size_t)bh * 2048;

  for (int j = 0; j < 32; ++j) {
    float gij = Gp[l * 32 + j], gji = Gp[j * 32 + l];
    Lm[l][j] = 0.5f * (gij + gji) * inv2 + ((l == j) ? RIDGE_C : 0.f);
    Mm[l][j] = Mp[l * 32 + j] * inv2;
    Cv[l][j]      = Cp[l * 32 + j] * inv;
    Cv[l + 32][j] = Cp[(l + 32) * 32 + j] * inv;
  }
  __syncthreads();

  // Cholesky (lower), in place on Lm
  for (int j = 0; j < 32; ++j) {
    if (l == j) {
      float s = Lm[j][j];
      for (int k = 0; k < j; ++k) s -= Lm[j][k] * Lm[j][k];
      Lm[j][j] = sqrtf(fmaxf(s, 1e-20f));
    }
    __syncthreads();
    if (l > j) {
      float s = Lm[l][j];
      for (int k = 0; k < j; ++k) s -= Lm[l][k] * Lm[j][k];
      Lm[l][j] = s / Lm[j][j];
    }
    __syncthreads();
  }
  for (int j = l + 1; j < 32; ++j) Lm[l][j] = 0.f;
  __syncthreads();

  // Li = L^-1 : lane owns column l (forward substitution)
  for (int r = 0; r < 32; ++r) {
    float s = (r == l) ? 1.f : 0.f;
    for (int k = l; k < r; ++k) s -= Lm[r][k] * Li[k][l];
    Li[r][l] = (r >= l) ? (s / Lm[r][r]) : 0.f;
  }
  __syncthreads();

  // Tt = M * Li^T ; Aw = Li * Tt  (lane owns column l)
  for (int i = 0; i < 32; ++i) {
    float s = 0.f;
    for (int k = 0; k < 32; ++k) s += Mm[i][k] * Li[l][k];
    Tt[i][l] = s;
  }
  __syncthreads();
  for (int i = 0; i < 32; ++i) {
    float s = 0.f;
    for (int k = 0; k < 32; ++k) s += Li[i][k] * Tt[k][l];
    Aw[i][l] = s;
  }
  __syncthreads();

  // power iteration: sigma_max(Aw)
  vec[l] = 0.17677669529663687f;  // 1/sqrt(32)
  __syncthreads();
  for (int it = 0; it < 6; ++it) {
    float s = 0.f;
    for (int j = 0; j < 32; ++j) s += Aw[l][j] * vec[j];
    red[l] = s * s; __syncthreads();
    for (int o = 16; o > 0; o >>= 1) { if (l < o) red[l] += red[l + o]; __syncthreads(); }
    float n1 = fmaxf(sqrtf(red[0]), 1e-8f); __syncthreads();
    uu[l] = s / n1; __syncthreads();
    float t = 0.f;
    for (int j = 0; j < 32; ++j) t += Aw[j][l] * uu[j];
    red[l] = t * t; __syncthreads();
    for (int o = 16; o > 0; o >>= 1) { if (l < o) red[l] += red[l + o]; __syncthreads(); }
    float n2 = fmaxf(sqrtf(red[0]), 1e-8f); __syncthreads();
    vec[l] = t / n2; __syncthreads();
  }
  float s = 0.f;
  for (int j = 0; j < 32; ++j) s += Aw[l][j] * vec[j];
  red[l] = s * s; __syncthreads();
  for (int o = 16; o > 0; o >>= 1) { if (l < o) red[l] += red[l + o]; __syncthreads(); }
  float sigma = sqrtf(red[0]); __syncthreads();

  float gamma = gamma_p[0];
  gamma = fminf(fmaxf(gamma, 1.0f), 1.5f);
  const float sca = gamma / fmaxf(sigma, 1.0f);
  for (int i = 0; i < 32; ++i) Aw[i][l] *= sca;
  __syncthreads();

  // Gi = Li^T Li (= G_sym^-1) ; Bv = Cv * Gi
  for (int i = 0; i < 32; ++i) {
    float t = 0.f;
    for (int k = 0; k < 32; ++k) t += Li[k][i] * Li[k][l];
    Gi[i][l] = t;
  }
  __syncthreads();
  for (int i = 0; i < 64; ++i) {
    float t = 0.f;
    for (int k = 0; k < 32; ++k) t += Cv[i][k] * Gi[k][l];
    Bv[i][l] = t;
  }
  __syncthreads();

  // S3 = L * Aw * Aw * Li ; E = Bv * S3 ; store E^T scaled by eta/maxnorm
  for (int i = 0; i < 32; ++i) {
    float t = 0.f;
    for (int k = 0; k < 32; ++k) t += Aw[i][k] * Li[k][l];
    S1[i][l] = t;
  }
  __syncthreads();
  for (int i = 0; i < 32; ++i) {
    float t = 0.f;
    for (int k = 0; k < 32; ++k) t += Aw[i][k] * S1[k][l];
    S2[i][l] = t;
  }
  __syncthreads();
  for (int i = 0; i < 32; ++i) {
    float t = 0.f;
    for (int k = 0; k < 32; ++k) t += Lm[i][k] * S2[k][l];
    S3[i][l] = t;
  }
  __syncthreads();
  const float fs = eta_p[0] * inv;
  float* Ep = Ert + (size_t)bh * 2048 + (size_t)l * 64;  // Ert[r=l][p]
  for (int i = 0; i < 64; ++i) {
    float t = 0.f;
    for (int k = 0; k < 32; ++k) t += Bv[i][k] * S3[k][l];
    Ep[i] = t * fs;
  }
}

// ---------------------------------------------------------------------------
// Kernel 5: y[b,t,h,p] = sum_r zq[b,t,h,r] * Ert[b,h,r,p]   (WMMA, K=32)
// Block: 256 thr; 8 waves x 16 tokens = 128 tokens, N=64.
// ---------------------------------------------------------------------------
__global__ __launch_bounds__(256)
void out_gemm_kernel(const float* __restrict__ zq, const float* __restrict__ Ert,
                     float* __restrict__ out) {
  __shared__ float sE[32][64];
  const int bh = blockIdx.x;
  const int b = bh >> 4, hh = bh & 15;
  const int t0 = blockIdx.y * 128;
  const int tid = threadIdx.x, lane = tid & 31, wave = tid >> 5;
  {
    const float* Ep = Ert + (size_t)bh * 2048;
    int base = tid * 8;
    float4 q0 = *(const float4*)(Ep + base);
    float4 q1 = *(const float4*)(Ep + base + 4);
    *(float4*)&((float*)sE)[base]     = q0;
    *(float4*)&((float*)sE)[base + 4] = q1;
  }
  __syncthreads();
  const int m = lane & 15, half = lane >> 4, kb = half * 2;
  const int tw = t0 + wave * 16;
  const float* zrow = zq + (size_t)b * T_DIM * 512 + (size_t)(tw + m) * 512 + hh * 32;
  v8f acc0 = {}, acc1 = {}, acc2 = {}, acc3 = {};
#pragma unroll
  for (int k0 = 0; k0 < 32; k0 += 4) {
    v2f a; a[0] = zrow[k0 + kb]; a[1] = zrow[k0 + kb + 1];
    v2f b0; b0[0] = sE[k0 + kb][m];      b0[1] = sE[k0 + kb + 1][m];
    v2f b1; b1[0] = sE[k0 + kb][16 + m]; b1[1] = sE[k0 + kb + 1][16 + m];
    v2f b2; b2[0] = sE[k0 + kb][32 + m]; b2[1] = sE[k0 + kb + 1][32 + m];
    v2f b3; b3[0] = sE[k0 + kb][48 + m]; b3[1] = sE[k0 + kb + 1][48 + m];
    acc0 = wmma4(a, b0, acc0);
    acc1 = wmma4(a, b1, acc1);
    acc2 = wmma4(a, b2, acc2);
    acc3 = wmma4(a, b3, acc3);
  }
  float* op = out + (size_t)b * T_DIM * 1024 + (size_t)hh * 64;
#pragma unroll
  for (int jv = 0; jv < 8; ++jv) {
    size_t trow = (size_t)(tw + jv + 8 * half);
    op[trow * 1024 + m]      = acc0[jv];
    op[trow * 1024 + 16 + m] = acc1[jv];
    op[trow * 1024 + 32 + m] = acc2[jv];
    op[trow * 1024 + 48 + m] = acc3[jv];
  }
}

// ---------------------------------------------------------------------------
extern "C" void kernel_launch(void* const* d_in, const int* in_sizes, int n_in,
                              void* d_out, int out_size, void* d_ws, size_t ws_size,
                              hipStream_t stream) {
  (void)in_sizes; (void)n_in; (void)out_size; (void)ws_size;
  const float* h   = (const float*)d_in[0];
  const float* pm  = (const float*)d_in[1];
  const float* Wk  = (const float*)d_in[2];
  const float* Wq  = (const float*)d_in[3];
  const float* Wv  = (const float*)d_in[4];
  const float* eta = (const float*)d_in[5];
  const float* gam = (const float*)d_in[6];

  char* ws = (char*)d_ws;
  const size_t MB = 1024ull * 1024ull;
  float* z    = (float*)(ws + 0);                       // 32 MB  (B,T,H*R)
  float* zq   = (float*)(ws + 32 * MB);                 // 32 MB
  float* v    = (float*)(ws + 64 * MB);                 // 64 MB  (B,T,H*P)
  float* G    = (float*)(ws + 128 * MB);                // 256 KB (B*H,32,32)
  float* M    = (float*)(ws + 128 * MB + 262144);       // 256 KB
  float* Cv   = (float*)(ws + 128 * MB + 524288);       // 512 KB (B*H,64,32)
  float* Ert  = (float*)(ws + 128 * MB + 1048576);      // 512 KB (B*H,32,64)
  float* imx  = (float*)(ws + 128 * MB + 1572864);      // 256 B  (B*H)

  qkv_gemm_kernel<<<dim3(32, 128), 256, 0, stream>>>(h, Wk, Wq, Wv, z, zq, v);
  maxnorm_kernel<<<64, 256, 0, stream>>>(z, imx);
  gram_kernel<<<64, 256, 0, stream>>>(z, v, pm, G, M, Cv);
  solve_kernel<<<64, 32, 0, stream>>>(G, M, Cv, imx, eta, gam, Ert);
  out_gemm_kernel<<<dim3(64, 32), 256, 0, stream>>>(zq, Ert, (float*)d_out);
}